// HybridGraphQLSTMNet_65481071399201
// MI455X (gfx1250) — compile-verified
//
#include <hip/hip_runtime.h>
#include <math.h>

typedef __attribute__((ext_vector_type(2))) float v2f;
typedef __attribute__((ext_vector_type(8))) float v8f;
typedef __attribute__((ext_vector_type(4))) int   v4i;

#if defined(__HIP_DEVICE_COMPILE__) && __has_builtin(__builtin_amdgcn_global_load_async_to_lds_b128)
#define QL_ASYNC 1
#else
#define QL_ASYNC 0
#endif

#if QL_ASYNC
typedef __attribute__((address_space(1))) v4i* ql_gvp;   // global int32x4*
typedef __attribute__((address_space(3))) v4i* ql_lvp;   // LDS    int32x4*
#endif

template <int N>
__device__ __forceinline__ void ql_wait_asynccnt() {
#if QL_ASYNC
    asm volatile("s_wait_asynccnt %0" ::"n"(N) : "memory");
#endif
}

// ---------------------------------------------------------------------------
// Phase 1: Zpre[t][n] = sum_{k<8} Wx[n][k]*latent[t][k] + b[n]
//   n = gate*4 + q, gates ordered {f,i,u,g}; Wx[n][k] = W_gate[q][k] (k<8)
//   One wave32 per 16-row tile; 2x V_WMMA_F32_16X16X4_F32 (K=8 split in two).
// ---------------------------------------------------------------------------
__global__ __launch_bounds__(32)
void qlstm_pre_kernel(const float* __restrict__ x,    // 4096x8 node_features
                      const float* __restrict__ Wf, const float* __restrict__ bf,
                      const float* __restrict__ Wi, const float* __restrict__ bi,
                      const float* __restrict__ Wu, const float* __restrict__ bu,
                      const float* __restrict__ Wg, const float* __restrict__ bg,
                      float* __restrict__ Zpre)       // 4096x16
{
    const int lane = threadIdx.x;            // 0..31
    const int tile = blockIdx.x;
    const int mrow = lane & 15;              // M within tile
    const int half = lane >> 4;              // 0: K={0,1}, 1: K={2,3}
    const int kb   = half * 2;
    const int row  = tile * 16 + mrow;

    // A fragment: latent values at K = kb, kb+1, kb+4, kb+5 (direct loads)
    const float* xr = x + row * 8;
    v2f a_lo, a_hi;
    a_lo.x = cosf(xr[kb + 0]);
    a_lo.y = cosf(xr[kb + 1]);
    a_lo.y *= (half == 0) ? a_lo.x : 1.f;    // latent col1 = cos(x0)*cos(x1)
    a_hi.x = cosf(xr[kb + 4]);
    a_hi.y = cosf(xr[kb + 5]);

    // B fragment: column n -> W_gate row q, first 8 columns
    const int n    = lane & 15;
    const int gate = n >> 2;
    const int q    = n & 3;
    const float* Wn = (gate == 0) ? Wf : (gate == 1) ? Wi : (gate == 2) ? Wu : Wg;
    Wn += q * 12;
    const float* bv = (gate == 0) ? bf : (gate == 1) ? bi : (gate == 2) ? bu : bg;
    const float bias = bv[q];

    v2f b_lo, b_hi;
    b_lo.x = Wn[kb + 0]; b_lo.y = Wn[kb + 1];
    b_hi.x = Wn[kb + 4]; b_hi.y = Wn[kb + 5];

    v8f acc = {};
    acc = __builtin_amdgcn_wmma_f32_16x16x4_f32(false, a_lo, false, b_lo,
                                                (short)0, acc, false, false);
    acc = __builtin_amdgcn_wmma_f32_16x16x4_f32(false, a_hi, false, b_hi,
                                                (short)0, acc, false, false);

#pragma unroll
    for (int v = 0; v < 8; ++v) {
        const int m = tile * 16 + v + half * 8;  // D row per ISA C/D layout
        Zpre[m * 16 + n] = acc[v] + bias;
    }
}

// ---------------------------------------------------------------------------
// Phase 2: the unavoidable serial LSTM scan (T steps), one wave32.
// Lane n (0..15) owns gate-output n; lanes 16..31 shadow 0..15 harmlessly.
// Zpre is double-buffered into LDS in 64-step (4KB) chunks via
// GLOBAL_LOAD_ASYNC_TO_LDS_B128, with s_wait_asynccnt for synchronization:
// the next chunk's DMA overlaps the current chunk's 64 serial steps.
// ---------------------------------------------------------------------------
#define QL_CHUNK 64                       // rows per chunk (4KB)
#define QL_CHUNK_F (QL_CHUNK * 16)        // floats per chunk
#define QL_LOADS 8                        // async b128 issues per chunk

__global__ __launch_bounds__(32)
void qlstm_scan_kernel(const float* __restrict__ Zpre,   // T x 16
                       const float* __restrict__ Wf, const float* __restrict__ thf,
                       const float* __restrict__ Wi, const float* __restrict__ thi,
                       const float* __restrict__ Wu, const float* __restrict__ thu,
                       const float* __restrict__ Wg, const float* __restrict__ thg,
                       float* __restrict__ Hout,         // T x 4
                       int T)
{
    __shared__ float zbuf[2 * QL_CHUNK_F];   // 8KB double buffer

    const int lane = threadIdx.x;
    const int n    = lane & 15;
    const int gate = n >> 2;
    const int q    = n & 3;

    const float* Wn = (gate == 0) ? Wf : (gate == 1) ? Wi : (gate == 2) ? Wu : Wg;
    Wn += q * 12;
    const float wh0 = Wn[8], wh1 = Wn[9], wh2 = Wn[10], wh3 = Wn[11];
    const float* thv = (gate == 0) ? thf : (gate == 1) ? thi : (gate == 2) ? thu : thg;
    const float th = thv[q];

    float h = 0.f, cst = 0.f;   // live in lanes 0..3 (q)

    // One LSTM step given this lane's pre-activation z at time t.
    auto step = [&](int t, float z) {
        const float h0 = __shfl(h, 0, 32);
        const float h1 = __shfl(h, 1, 32);
        const float h2 = __shfl(h, 2, 32);
        const float h3 = __shfl(h, 3, 32);
        z += wh0 * h0 + wh1 * h1 + wh2 * h2 + wh3 * h3;

        const float c = cosf(z + th);
        const int base = gate * 4;
        const float c0 = __shfl(c, base + 0, 32);
        const float c1 = __shfl(c, base + 1, 32);
        const float c2 = __shfl(c, base + 2, 32);
        const float c3 = __shfl(c, base + 3, 32);

        // Z-expectations from the CNOT-ring parity analysis
        float e;
        if      (q == 0) e = c1 * c2 * c3;
        else if (q == 1) e = c0 * c1;
        else if (q == 2) e = c0 * c1 * c2;
        else             e = c0 * c1 * c2 * c3;

        // f,i,o -> sigmoid ; u -> tanh
        const float v = (gate == 2) ? tanhf(e) : (1.f / (1.f + expf(-e)));

        const float fq = __shfl(v, q,      32);
        const float iq = __shfl(v, 4 + q,  32);
        const float gq = __shfl(v, 8 + q,  32);
        const float oq = __shfl(v, 12 + q, 32);
        cst = fq * cst + iq * gq;
        h   = oq * tanhf(cst);

        if (lane < 4) Hout[t * 4 + q] = h;
    };

#if QL_ASYNC
    auto issue_chunk = [&](int chunk, int buf) {
        const float* gsrc = Zpre + (size_t)chunk * QL_CHUNK_F;
        float*       ldst = &zbuf[buf * QL_CHUNK_F];
#pragma unroll
        for (int i = 0; i < QL_LOADS; ++i) {
            __builtin_amdgcn_global_load_async_to_lds_b128(
                (ql_gvp)(gsrc + i * 128 + lane * 4),
                (ql_lvp)(ldst + i * 128 + lane * 4),
                /*offset=*/0, /*cpol=*/0);
        }
    };

    const int chunks = T / QL_CHUNK;
    if (chunks > 0) issue_chunk(0, 0);
    for (int cpos = 0; cpos < chunks; ++cpos) {
        const int buf = cpos & 1;
        if (cpos + 1 < chunks) {
            issue_chunk(cpos + 1, buf ^ 1);
            ql_wait_asynccnt<QL_LOADS>();   // current chunk landed
        } else {
            ql_wait_asynccnt<0>();
        }
        const float* zrow = &zbuf[buf * QL_CHUNK_F];
        for (int j = 0; j < QL_CHUNK; ++j)
            step(cpos * QL_CHUNK + j, zrow[j * 16 + n]);
    }
    for (int t = chunks * QL_CHUNK; t < T; ++t)   // tail (T not multiple of 64)
        step(t, Zpre[t * 16 + n]);
#else
    for (int t = 0; t < T; ++t) {
        const int tp = (t + 32 < T) ? (t + 32) : t;
        __builtin_prefetch(Zpre + tp * 16, 0, 1);
        step(t, Zpre[t * 16 + n]);
    }
#endif
}

// ---------------------------------------------------------------------------
// Phase 3: out = H(Tx4) @ Wout^T(4x8) + bout, one WMMA (K=4) per 16-row
// tile, N padded 8->16 with zero columns; masked store after the WMMA.
// ---------------------------------------------------------------------------
__global__ __launch_bounds__(32)
void qlstm_out_kernel(const float* __restrict__ H,      // T x 4
                      const float* __restrict__ Wout,   // 8x4
                      const float* __restrict__ bout,   // 8
                      float* __restrict__ out)          // T x 8
{
    const int lane = threadIdx.x;
    const int tile = blockIdx.x;
    const int mrow = lane & 15;
    const int half = lane >> 4;
    const int kb   = half * 2;
    const int row  = tile * 16 + mrow;
    const int n    = lane & 15;

    const float mask = (n < 8) ? 1.f : 0.f;
    const int   nc   = n & 7;               // in-bounds column index

    v2f a, b;
    a.x = H[row * 4 + kb + 0];
    a.y = H[row * 4 + kb + 1];
    b.x = Wout[nc * 4 + kb + 0] * mask;
    b.y = Wout[nc * 4 + kb + 1] * mask;

    v8f acc = {};
    acc = __builtin_amdgcn_wmma_f32_16x16x4_f32(false, a, false, b,
                                                (short)0, acc, false, false);

    if (n < 8) {
        const float bias = bout[n];
#pragma unroll
        for (int v = 0; v < 8; ++v) {
            const int m = tile * 16 + v + half * 8;
            out[m * 8 + n] = acc[v] + bias;
        }
    }
}

// ---------------------------------------------------------------------------
extern "C" void kernel_launch(void* const* d_in, const int* in_sizes, int n_in,
                              void* d_out, int out_size, void* d_ws, size_t ws_size,
                              hipStream_t stream) {
    const float* x    = (const float*)d_in[0];   // node_features (N,8)
    // d_in[1] = adjacency: unused by the reference forward pass
    const float* Wf = (const float*)d_in[2];
    const float* bf = (const float*)d_in[3];
    const float* thf = (const float*)d_in[4];
    const float* Wi = (const float*)d_in[5];
    const float* bi = (const float*)d_in[6];
    const float* thi = (const float*)d_in[7];
    const float* Wu = (const float*)d_in[8];
    const float* bu = (const float*)d_in[9];
    const float* thu = (const float*)d_in[10];
    const float* Wg = (const float*)d_in[11];
    const float* bg = (const float*)d_in[12];
    const float* thg = (const float*)d_in[13];
    const float* Wout = (const float*)d_in[14];
    const float* bout = (const float*)d_in[15];
    float* out = (float*)d_out;

    const int N     = in_sizes[0] / 8;   // 4096
    const int tiles = N / 16;            // 256

    float* Zpre = (float*)d_ws;          // N*16 floats
    float* Hout = Zpre + (size_t)N * 16; // N*4  floats

    qlstm_pre_kernel<<<tiles, 32, 0, stream>>>(x, Wf, bf, Wi, bi, Wu, bu, Wg, bg, Zpre);
    qlstm_scan_kernel<<<1, 32, 0, stream>>>(Zpre, Wf, thf, Wi, thi, Wu, thu, Wg, thg, Hout, N);
    qlstm_out_kernel<<<tiles, 32, 0, stream>>>(Hout, Wout, bout, out);
}